// SDRLoss_62680752717874
// MI455X (gfx1250) — compile-verified
//
#include <hip/hip_runtime.h>
#include <hip/hip_bf16.h>

#define T_LEN 128000
#define N_CH  64          // B*C = 32*2
#define L_LAG 512
#define EPSF  1e-8f
#define SPLIT 2           // workgroups per channel in corr kernel

typedef float v2f __attribute__((ext_vector_type(2)));
typedef float v8f __attribute__((ext_vector_type(8)));

// ---------------------------------------------------------------------------
// Kernel 1: per-signal mean and inverse norm of zero-meaned signal.
// 128 signals: s<64 -> est channel s ; s>=64 -> ref channel s-64.
// ---------------------------------------------------------------------------
__global__ __launch_bounds__(256) void sdr_stats_kernel(
    const float* __restrict__ est, const float* __restrict__ ref,
    float* __restrict__ stats) {
  const int s = blockIdx.x;                // 0..127
  const float* src = (s < N_CH ? est : ref) + (size_t)(s & (N_CH - 1)) * T_LEN;
  float sum = 0.f, ss = 0.f;
  for (int t = threadIdx.x; t < T_LEN; t += 256) {
    float v = src[t];
    sum += v;
    ss  += v * v;
  }
  for (int off = 16; off; off >>= 1) {
    sum += __shfl_xor(sum, off, 32);
    ss  += __shfl_xor(ss,  off, 32);
  }
  __shared__ float red[16];
  const int warp = threadIdx.x >> 5;
  if ((threadIdx.x & 31) == 0) { red[warp] = sum; red[8 + warp] = ss; }
  __syncthreads();
  if (threadIdx.x == 0) {
    sum = 0.f; ss = 0.f;
    for (int w = 0; w < 8; ++w) { sum += red[w]; ss += red[8 + w]; }
    const float mean = sum / (float)T_LEN;
    const float en   = fmaxf(ss - sum * mean, 0.f);   // ||x - mean||^2
    stats[2 * s]     = mean;
    stats[2 * s + 1] = 1.f / (sqrtf(en) + EPSF);
  }
}

// ---------------------------------------------------------------------------
// Corr helpers. Normalization folded into one FMA: v = x*sc + bi,
// with sc = inv_norm, bi = -mean*inv_norm.
// ---------------------------------------------------------------------------
__device__ __forceinline__ float ldn_clamped(const float* __restrict__ p,
                                             int t, float sc, float bi) {
  int tc = t < 0 ? 0 : (t > T_LEN - 1 ? T_LEN - 1 : t);
  float v = fmaf(p[tc], sc, bi);          // always-safe load + cndmask
  return (tc == t) ? v : 0.f;
}

#define CORR_WMMA4()                                                          \
  accA0 = __builtin_amdgcn_wmma_f32_16x16x4_f32(false, A0, false, Br,         \
                                                (short)0, accA0, false, false);\
  accA1 = __builtin_amdgcn_wmma_f32_16x16x4_f32(false, A1, false, Br,         \
                                                (short)0, accA1, false, false);\
  accX0 = __builtin_amdgcn_wmma_f32_16x16x4_f32(false, A0, false, Be,         \
                                                (short)0, accX0, false, false);\
  accX1 = __builtin_amdgcn_wmma_f32_16x16x4_f32(false, A1, false, Be,         \
                                                (short)0, accX1, false, false)

// ---------------------------------------------------------------------------
// Kernel 2: acf/xcorr via fp32 WMMA (GEMM form):
//   out[a][b] = sum_u A[a][u]*B[u][b], A[a][u]=refn[u-16a], B[u][b]=sign[u+b]
//   lag k = 16a+b ; a in 0..31 (two 16-row tiles), b in 0..15.
// SPLIT workgroups per channel; 8 waves each; per-wave u-slice accumulated
// with v_wmma_f32_16x16x4_f32; block partials reduced through LDS.
// ---------------------------------------------------------------------------
__global__ __launch_bounds__(256) void sdr_corr_kernel(
    const float* __restrict__ est, const float* __restrict__ ref,
    const float* __restrict__ stats,
    float* __restrict__ acfP, float* __restrict__ xcrP) {
  __shared__ float lds[8 * 4 * 256];   // 8 waves * 4 tiles * 256 f32 = 32KB
  const int bid  = blockIdx.x;
  const int ch   = bid >> 1;           // SPLIT == 2
  const int part = bid & 1;
  const float* eb = est + (size_t)ch * T_LEN;
  const float* rb = ref + (size_t)ch * T_LEN;
  const float ie = stats[2 * ch + 1];
  const float be_ = -stats[2 * ch] * ie;
  const float ir = stats[2 * (N_CH + ch) + 1];
  const float br_ = -stats[2 * (N_CH + ch)] * ir;

  const int tid  = threadIdx.x;
  const int w    = tid >> 5;
  const int lane = tid & 31;
  const int m    = lane & 15;        // A row within tile / B column
  const int kk   = (lane >> 4) * 2;  // lane-half selects K pair {0,1} vs {2,3}

  v8f accA0 = {}, accA1 = {}, accX0 = {}, accX1 = {};

  // wave slice: 16 slices of 8000 u-values
  const int si  = part * 8 + w;
  const int uLo = si * (T_LEN / (8 * SPLIT));
  const int uHi = uLo + (T_LEN / (8 * SPLIT));
  int uA = uLo < 496 ? 496 : uLo;            if (uA > uHi) uA = uHi;
  int uB = uHi > T_LEN - 20 ? T_LEN - 20 : uHi; if (uB < uA) uB = uA;

  // ---- leading edge (A may index t<0): clamped loads, branch-free ----
  for (int u = uLo; u < uA; u += 4) {
    const int t0 = u + kk - 16 * m;
    const int tb = u + kk + m;
    v2f A0, A1, Br, Be;
    A0[0] = ldn_clamped(rb, t0,       ir, br_);
    A0[1] = ldn_clamped(rb, t0 + 1,   ir, br_);
    A1[0] = ldn_clamped(rb, t0 - 256, ir, br_);
    A1[1] = ldn_clamped(rb, t0 - 255, ir, br_);
    Br[0] = ldn_clamped(rb, tb,       ir, br_);
    Br[1] = ldn_clamped(rb, tb + 1,   ir, br_);
    Be[0] = ldn_clamped(eb, tb,       ie, be_);
    Be[1] = ldn_clamped(eb, tb + 1,   ie, be_);
    CORR_WMMA4();
  }

  // ---- fast interior: no bounds checks, vector A loads ----
  #pragma unroll 2
  for (int u = uA; u < uB; u += 4) {
    const int t0 = u + kk - 16 * m;       // even -> 8B-aligned float2
    const int tb = u + kk + m;
    v2f a0 = *(const v2f*)(rb + t0);
    v2f a1 = *(const v2f*)(rb + t0 - 256);
    float r0 = rb[tb], r1 = rb[tb + 1];
    float e0 = eb[tb], e1 = eb[tb + 1];
    v2f A0, A1, Br, Be;
    A0[0] = fmaf(a0[0], ir, br_); A0[1] = fmaf(a0[1], ir, br_);
    A1[0] = fmaf(a1[0], ir, br_); A1[1] = fmaf(a1[1], ir, br_);
    Br[0] = fmaf(r0, ir, br_);    Br[1] = fmaf(r1, ir, br_);
    Be[0] = fmaf(e0, ie, be_);    Be[1] = fmaf(e1, ie, be_);
    CORR_WMMA4();
  }

  // ---- trailing edge (B may index t>=T): clamped loads ----
  for (int u = uB; u < uHi; u += 4) {
    const int t0 = u + kk - 16 * m;
    const int tb = u + kk + m;
    v2f A0, A1, Br, Be;
    A0[0] = ldn_clamped(rb, t0,       ir, br_);
    A0[1] = ldn_clamped(rb, t0 + 1,   ir, br_);
    A1[0] = ldn_clamped(rb, t0 - 256, ir, br_);
    A1[1] = ldn_clamped(rb, t0 - 255, ir, br_);
    Br[0] = ldn_clamped(rb, tb,       ir, br_);
    Br[1] = ldn_clamped(rb, tb + 1,   ir, br_);
    Be[0] = ldn_clamped(eb, tb,       ie, be_);
    Be[1] = ldn_clamped(eb, tb + 1,   ie, be_);
    CORR_WMMA4();
  }

  // park per-wave partial tiles in LDS
  #pragma unroll
  for (int r = 0; r < 8; ++r) {
    lds[(w * 4 + 0) * 256 + lane * 8 + r] = accA0[r];
    lds[(w * 4 + 1) * 256 + lane * 8 + r] = accA1[r];
    lds[(w * 4 + 2) * 256 + lane * 8 + r] = accX0[r];
    lds[(w * 4 + 3) * 256 + lane * 8 + r] = accX1[r];
  }
  __syncthreads();

  // reduce across 8 waves; C/D layout: VGPR r, lanes0-15 -> M=r, lanes16-31 -> M=r+8
  float* acfO = acfP + (size_t)(part * N_CH + ch) * L_LAG;
  float* xcrO = xcrP + (size_t)(part * N_CH + ch) * L_LAG;
  for (int lag = tid; lag < L_LAG; lag += 256) {
    const int atile = lag >> 8, rem = lag & 255;
    const int mm = rem >> 4, nn = rem & 15;
    const int rr = mm & 7, ln = (mm >> 3) * 16 + nn;
    float sa = 0.f, sx = 0.f;
    for (int ww = 0; ww < 8; ++ww) {
      sa += lds[(ww * 4 + 0 + atile) * 256 + ln * 8 + rr];
      sx += lds[(ww * 4 + 2 + atile) * 256 + ln * 8 + rr];
    }
    acfO[lag] = sa;
    xcrO[lag] = sx;
  }
}

// ---------------------------------------------------------------------------
// Kernel 3: Levinson-Durbin solve of R h = xcorr (symmetric Toeplitz),
// then coherence -> negative SDR per channel. One block per channel.
// Sums the SPLIT partials while loading.
// ---------------------------------------------------------------------------
__global__ __launch_bounds__(256) void sdr_levinson_kernel(
    const float* __restrict__ acfP, const float* __restrict__ xcrP,
    float* __restrict__ sdr) {
  __shared__ float r[L_LAG], y[L_LAG], f[L_LAG], x[L_LAG];
  __shared__ float red[16];
  const int ch = blockIdx.x;
  const int tid = threadIdx.x;
  const size_t stride = (size_t)N_CH * L_LAG;
  for (int i = tid; i < L_LAG; i += 256) {
    r[i] = acfP[(size_t)ch * L_LAG + i] + acfP[stride + (size_t)ch * L_LAG + i];
    y[i] = xcrP[(size_t)ch * L_LAG + i] + xcrP[stride + (size_t)ch * L_LAG + i];
    f[i] = 0.f; x[i] = 0.f;
  }
  __syncthreads();
  if (tid == 0) {
    const float inv0 = 1.f / fmaxf(r[0], 1e-20f);
    f[0] = inv0;
    x[0] = y[0] * inv0;
  }
  __syncthreads();

  for (int n = 1; n < L_LAG; ++n) {
    float ef = 0.f, ex = 0.f;
    for (int i = tid; i < n; i += 256) {
      const float rn = r[n - i];
      ef += rn * f[i];
      ex += rn * x[i];
    }
    for (int off = 16; off; off >>= 1) {
      ef += __shfl_xor(ef, off, 32);
      ex += __shfl_xor(ex, off, 32);
    }
    const int warp = tid >> 5;
    if ((tid & 31) == 0) { red[warp] = ef; red[8 + warp] = ex; }
    __syncthreads();
    ef = 0.f; ex = 0.f;
    for (int ww = 0; ww < 8; ++ww) { ef += red[ww]; ex += red[8 + ww]; }
    const float alpha = 1.f / fmaxf(1.f - ef * ef, 1e-20f);
    const float beta  = y[n] - ex;

    float nf[2], nx[2]; int is[2]; int cnt = 0;
    for (int i = tid; i <= n; i += 256) {
      const float fi = f[i];          // f[n] == 0 pre-update
      const float bi = f[n - i];
      nf[cnt] = alpha * (fi - ef * bi);
      nx[cnt] = x[i] + beta * alpha * (bi - ef * fi);
      is[cnt] = i; ++cnt;
    }
    __syncthreads();
    for (int c = 0; c < cnt; ++c) { f[is[c]] = nf[c]; x[is[c]] = nx[c]; }
    __syncthreads();
  }

  float coh = 0.f;
  for (int i = tid; i < L_LAG; i += 256) coh += y[i] * x[i];
  for (int off = 16; off; off >>= 1) coh += __shfl_xor(coh, off, 32);
  if ((tid & 31) == 0) red[tid >> 5] = coh;
  __syncthreads();
  if (tid == 0) {
    coh = 0.f;
    for (int ww = 0; ww < 8; ++ww) coh += red[ww];
    const float ratio = coh / fmaxf(1.f - coh, EPSF);
    sdr[ch] = -10.f * log10f(fmaxf(ratio, EPSF));
  }
}

// ---------------------------------------------------------------------------
// Kernel 4: mean over 64 channels -> scalar.
// ---------------------------------------------------------------------------
__global__ __launch_bounds__(64) void sdr_finalize_kernel(
    const float* __restrict__ sdr, float* __restrict__ out) {
  const int tid = threadIdx.x;
  float v = sdr[tid];
  for (int off = 16; off; off >>= 1) v += __shfl_xor(v, off, 32);
  __shared__ float tmp[2];
  if ((tid & 31) == 0) tmp[tid >> 5] = v;
  __syncthreads();
  if (tid == 0) out[0] = (tmp[0] + tmp[1]) * (1.0f / (float)N_CH);
}

// ---------------------------------------------------------------------------
extern "C" void kernel_launch(void* const* d_in, const int* in_sizes, int n_in,
                              void* d_out, int out_size, void* d_ws, size_t ws_size,
                              hipStream_t stream) {
  const float* est = (const float*)d_in[0];
  const float* ref = (const float*)d_in[1];
  float* out = (float*)d_out;
  float* ws  = (float*)d_ws;

  float* stats = ws;                                   // 256 floats
  float* acfP  = ws + 256;                             // SPLIT*64*512
  float* xcrP  = ws + 256 + SPLIT * N_CH * L_LAG;      // SPLIT*64*512
  float* sdr   = ws + 256 + 2 * SPLIT * N_CH * L_LAG;  // 64
  (void)in_sizes; (void)n_in; (void)out_size; (void)ws_size;

  sdr_stats_kernel   <<<2 * N_CH,     256, 0, stream>>>(est, ref, stats);
  sdr_corr_kernel    <<<SPLIT * N_CH, 256, 0, stream>>>(est, ref, stats, acfP, xcrP);
  sdr_levinson_kernel<<<N_CH,         256, 0, stream>>>(acfP, xcrP, sdr);
  sdr_finalize_kernel<<<1,             64, 0, stream>>>(sdr, out);
}